// KernelInterpolationFeatPAM_3934190044193
// MI455X (gfx1250) — compile-verified
//
#include <hip/hip_runtime.h>
#include <hip/hip_bf16.h>
#include <math.h>

typedef __attribute__((ext_vector_type(2))) float v2f;
typedef __attribute__((ext_vector_type(8))) float v8f;

#define MVOX 110592   // 48*48*48
#define NKP  1024
#define GDIM 48
#define GSL  2304     // 48*48

// ---------------------------------------------------------------------------
// Kernel 1: conv3d 19ch -> 19ch (input = feats[16] ++ analytic grid[3]), +bias,
// +ReLU. Weights staged in LDS (9747 floats = 39KB). One thread per voxel,
// all 19 output channels in registers.
// ---------------------------------------------------------------------------
__global__ void __launch_bounds__(256)
conv0_relu_kernel(const float* __restrict__ feats,
                  const float* __restrict__ w0,
                  const float* __restrict__ b0,
                  float* __restrict__ h0)
{
    __shared__ float sw[19 * 19 * 27];
    for (int i = threadIdx.x; i < 19 * 19 * 27; i += 256) sw[i] = w0[i];
    __syncthreads();

    int idx = blockIdx.x * 256 + threadIdx.x;
    if (idx >= MVOX) return;
    int z = idx / GSL;
    int y = (idx / GDIM) % GDIM;
    int x = idx % GDIM;

    const float gs = 2.0f / 47.0f;

    float acc[19];
    #pragma unroll
    for (int oc = 0; oc < 19; ++oc) acc[oc] = b0[oc];

    for (int dz = -1; dz <= 1; ++dz) {
        int zz = z + dz; if (zz < 0 || zz >= GDIM) continue;
        for (int dy = -1; dy <= 1; ++dy) {
            int yy = y + dy; if (yy < 0 || yy >= GDIM) continue;
            for (int dx = -1; dx <= 1; ++dx) {
                int xx = x + dx; if (xx < 0 || xx >= GDIM) continue;
                int tap  = (dz + 1) * 9 + (dy + 1) * 3 + (dx + 1);
                int base = zz * GSL + yy * GDIM + xx;
                #pragma unroll 1
                for (int ic = 0; ic < 19; ++ic) {
                    float val;
                    if (ic < 16)       val = feats[ic * MVOX + base];
                    else if (ic == 16) val = -1.0f + xx * gs;
                    else if (ic == 17) val = -1.0f + yy * gs;
                    else               val = -1.0f + zz * gs;
                    #pragma unroll
                    for (int oc = 0; oc < 19; ++oc)
                        acc[oc] = fmaf(val, sw[(oc * 19 + ic) * 27 + tap], acc[oc]);
                }
            }
        }
    }
    #pragma unroll
    for (int oc = 0; oc < 19; ++oc)
        h0[(size_t)oc * MVOX + idx] = fmaxf(acc[oc], 0.0f);
}

// ---------------------------------------------------------------------------
// Kernel 2: per-channel mean / invstd. One block per channel.
// ---------------------------------------------------------------------------
__global__ void __launch_bounds__(256)
in_stats_kernel(const float* __restrict__ x, float* __restrict__ st)
{
    int c = blockIdx.x;
    const float* xc = x + (size_t)c * MVOX;
    float s = 0.0f, sq = 0.0f;
    for (int i = threadIdx.x; i < MVOX; i += 256) {
        float v = xc[i];
        s += v; sq += v * v;
    }
    __shared__ float ss[256], ssq[256];
    ss[threadIdx.x] = s; ssq[threadIdx.x] = sq;
    __syncthreads();
    for (int off = 128; off > 0; off >>= 1) {
        if (threadIdx.x < off) {
            ss[threadIdx.x]  += ss[threadIdx.x + off];
            ssq[threadIdx.x] += ssq[threadIdx.x + off];
        }
        __syncthreads();
    }
    if (threadIdx.x == 0) {
        float mu  = ss[0] / (float)MVOX;
        float var = ssq[0] / (float)MVOX - mu * mu;
        st[c * 2 + 0] = mu;
        st[c * 2 + 1] = rsqrtf(var + 1e-5f);
    }
}

// Kernel 3: apply instance norm in place (19 channels)
__global__ void __launch_bounds__(256)
in_apply_kernel(float* __restrict__ x, const float* __restrict__ st)
{
    int i = blockIdx.x * 256 + threadIdx.x;
    if (i >= 19 * MVOX) return;
    int c = i / MVOX;
    x[i] = (x[i] - st[c * 2]) * st[c * 2 + 1];
}

// ---------------------------------------------------------------------------
// Kernel 4: conv3d 19ch -> 8ch + bias + ReLU. Weights in LDS (4104 floats).
// ---------------------------------------------------------------------------
__global__ void __launch_bounds__(256)
conv1_relu_kernel(const float* __restrict__ h0,
                  const float* __restrict__ w1,
                  const float* __restrict__ b1,
                  float* __restrict__ h1)
{
    __shared__ float sw[8 * 19 * 27];
    for (int i = threadIdx.x; i < 8 * 19 * 27; i += 256) sw[i] = w1[i];
    __syncthreads();

    int idx = blockIdx.x * 256 + threadIdx.x;
    if (idx >= MVOX) return;
    int z = idx / GSL;
    int y = (idx / GDIM) % GDIM;
    int x = idx % GDIM;

    float acc[8];
    #pragma unroll
    for (int oc = 0; oc < 8; ++oc) acc[oc] = b1[oc];

    for (int dz = -1; dz <= 1; ++dz) {
        int zz = z + dz; if (zz < 0 || zz >= GDIM) continue;
        for (int dy = -1; dy <= 1; ++dy) {
            int yy = y + dy; if (yy < 0 || yy >= GDIM) continue;
            for (int dx = -1; dx <= 1; ++dx) {
                int xx = x + dx; if (xx < 0 || xx >= GDIM) continue;
                int tap  = (dz + 1) * 9 + (dy + 1) * 3 + (dx + 1);
                int base = zz * GSL + yy * GDIM + xx;
                #pragma unroll 1
                for (int ic = 0; ic < 19; ++ic) {
                    float val = h0[(size_t)ic * MVOX + base];
                    #pragma unroll
                    for (int oc = 0; oc < 8; ++oc)
                        acc[oc] = fmaf(val, sw[(oc * 19 + ic) * 27 + tap], acc[oc]);
                }
            }
        }
    }
    #pragma unroll
    for (int oc = 0; oc < 8; ++oc)
        h1[(size_t)oc * MVOX + idx] = fmaxf(acc[oc], 0.0f);
}

// Kernel 5: apply instance norm to h1 (8 ch) in place + per-voxel |fg|^2
__global__ void __launch_bounds__(256)
in_apply_fgnorm_kernel(float* __restrict__ h1, const float* __restrict__ st,
                       float* __restrict__ fgn)
{
    int m = blockIdx.x * 256 + threadIdx.x;
    if (m >= MVOX) return;
    float s = 0.0f;
    #pragma unroll
    for (int c = 0; c < 8; ++c) {
        float v = (h1[(size_t)c * MVOX + m] - st[c * 2]) * st[c * 2 + 1];
        h1[(size_t)c * MVOX + m] = v;
        s += v * v;
    }
    fgn[m] = s;
}

// ---------------------------------------------------------------------------
// Kernel 6: trilinear grid-sample of the 8-ch field at 1024 keypoints.
// fk stored channel-major [8][1024] (= B-fragment friendly), plus |fk|^2.
// ---------------------------------------------------------------------------
__global__ void __launch_bounds__(256)
sample_kernel(const float* __restrict__ h, const float* __restrict__ kpts,
              float* __restrict__ fk, float* __restrict__ fkn)
{
    int n = blockIdx.x * 256 + threadIdx.x;
    if (n >= NKP) return;
    float ix = (kpts[n * 3 + 0] + 1.0f) * 0.5f * 47.0f;
    float iy = (kpts[n * 3 + 1] + 1.0f) * 0.5f * 47.0f;
    float iz = (kpts[n * 3 + 2] + 1.0f) * 0.5f * 47.0f;
    float xf = floorf(ix), yf = floorf(iy), zf = floorf(iz);
    float wx = ix - xf, wy = iy - yf, wz = iz - zf;
    int x0 = min(max((int)xf, 0), 47); int x1 = min(x0 + 1, 47);
    int y0 = min(max((int)yf, 0), 47); int y1 = min(y0 + 1, 47);
    int z0 = min(max((int)zf, 0), 47); int z1 = min(z0 + 1, 47);

    int b000 = z0 * GSL + y0 * GDIM + x0, b001 = z0 * GSL + y0 * GDIM + x1;
    int b010 = z0 * GSL + y1 * GDIM + x0, b011 = z0 * GSL + y1 * GDIM + x1;
    int b100 = z1 * GSL + y0 * GDIM + x0, b101 = z1 * GSL + y0 * GDIM + x1;
    int b110 = z1 * GSL + y1 * GDIM + x0, b111 = z1 * GSL + y1 * GDIM + x1;

    float s = 0.0f;
    #pragma unroll
    for (int c = 0; c < 8; ++c) {
        const float* hc = h + (size_t)c * MVOX;
        float c00 = hc[b000] * (1.0f - wx) + hc[b001] * wx;
        float c01 = hc[b010] * (1.0f - wx) + hc[b011] * wx;
        float c10 = hc[b100] * (1.0f - wx) + hc[b101] * wx;
        float c11 = hc[b110] * (1.0f - wx) + hc[b111] * wx;
        float c0  = c00 * (1.0f - wy) + c01 * wy;
        float c1  = c10 * (1.0f - wy) + c11 * wy;
        float v   = c0 * (1.0f - wz) + c1 * wz;
        fk[c * NKP + n] = v;
        s += v * v;
    }
    fkn[n] = s;
}

// ---------------------------------------------------------------------------
// Kernel 7 (main): fused RBF interpolation.
//   d2[m][n] = |fg_m|^2 + |fk_n|^2 - 2 * <fg_m, fk_n>   (K=8 GEMM via WMMA)
//   w = exp(-0.5 d2);  out[:,m] = (w @ disp) / (row-sum + 1e-8)
// One wave per 16-row M-tile; 64 N-tiles of 16 streamed from LDS.
// Dot products: two V_WMMA_F32_16X16X4_F32 per tile (K=8 = 2x4, full f32).
// Exponent folded to exp2 form: exp(-0.5 d2) = exp2(prn[r] + pf + c*log2e),
// prn hoisted out of the n-loop, pf one mul per column -> one add + one fma
// per element feeding v_exp_f32 directly (removes two pk_mul stages/iter).
// All waves fully active -> EXEC == all-ones at every WMMA.
// ---------------------------------------------------------------------------
__global__ void __launch_bounds__(256)
rbf_wmma_kernel(const float* __restrict__ fg,   // [8][MVOX] (== normalized h1)
                const float* __restrict__ fgn,  // [MVOX]
                const float* __restrict__ fk,   // [8][NKP]
                const float* __restrict__ fkn,  // [NKP]
                const float* __restrict__ disp, // [NKP][3]
                float* __restrict__ out)        // [3][MVOX]
{
    __shared__ float sfk[8 * NKP];   // 32 KB
    __shared__ float sfkn[NKP];      // 4 KB (pre-scaled by -0.5*log2e)
    __shared__ float sdisp[3 * NKP]; // 12 KB, transposed to [j][n]
    const float L2E  = 1.4426950408889634f;
    const float NHL2 = -0.5f * L2E;
    for (int i = threadIdx.x; i < 8 * NKP; i += 256) sfk[i] = fk[i];
    for (int i = threadIdx.x; i < NKP; i += 256) sfkn[i] = fkn[i] * NHL2;
    for (int i = threadIdx.x; i < 3 * NKP; i += 256)
        sdisp[(i % 3) * NKP + (i / 3)] = disp[i];
    __syncthreads();

    int wave = (blockIdx.x << 3) + (threadIdx.x >> 5); // 8 waves / block
    int lane = threadIdx.x & 31;
    int half = lane >> 4;          // 0: rows m0..m0+7 / K 0,1 ; 1: rows +8 / K 2,3
    int lm   = lane & 15;
    int m0   = wave << 4;          // grid covers MVOX/16 waves exactly
    int kb   = half * 2;

    // A fragments (reused across all 64 N-tiles)
    int row = m0 + lm;
    v2f a0, a1;
    a0.x = fg[(size_t)(kb + 0) * MVOX + row];
    a0.y = fg[(size_t)(kb + 1) * MVOX + row];
    a1.x = fg[(size_t)(kb + 4) * MVOX + row];
    a1.y = fg[(size_t)(kb + 5) * MVOX + row];

    int mrow = m0 + half * 8;
    float prn[8];                  // -0.5*log2e * |fg_row|^2, hoisted
    #pragma unroll
    for (int r = 0; r < 8; ++r) prn[r] = fgn[mrow + r] * NHL2;

    float ax[8], ay[8], az[8], aw[8];
    #pragma unroll
    for (int r = 0; r < 8; ++r) { ax[r] = 0.f; ay[r] = 0.f; az[r] = 0.f; aw[r] = 0.f; }

    #pragma unroll 1
    for (int n0 = 0; n0 < NKP; n0 += 16) {
        int col = n0 + lm;
        v2f b0v, b1v;
        b0v.x = sfk[(kb + 0) * NKP + col];
        b0v.y = sfk[(kb + 1) * NKP + col];
        b1v.x = sfk[(kb + 4) * NKP + col];
        b1v.y = sfk[(kb + 5) * NKP + col];

        v8f c = {0.f, 0.f, 0.f, 0.f, 0.f, 0.f, 0.f, 0.f};
        c = __builtin_amdgcn_wmma_f32_16x16x4_f32(false, a0, false, b0v,
                                                  (short)0, c, false, false);
        c = __builtin_amdgcn_wmma_f32_16x16x4_f32(false, a1, false, b1v,
                                                  (short)0, c, false, false);

        float pf = sfkn[col];      // already * -0.5*log2e
        float dx = sdisp[col], dy = sdisp[NKP + col], dzv = sdisp[2 * NKP + col];
        #pragma unroll
        for (int r = 0; r < 8; ++r) {
            float w = exp2f(fmaf(c[r], L2E, prn[r] + pf));
            ax[r] += w * dx;
            ay[r] += w * dy;
            az[r] += w * dzv;
            aw[r] += w;
        }
    }

    // reduce across the 16 lanes of each half (xor keeps halves separate)
    #pragma unroll
    for (int r = 0; r < 8; ++r) {
        #pragma unroll
        for (int off = 1; off < 16; off <<= 1) {
            ax[r] += __shfl_xor(ax[r], off, 32);
            ay[r] += __shfl_xor(ay[r], off, 32);
            az[r] += __shfl_xor(az[r], off, 32);
            aw[r] += __shfl_xor(aw[r], off, 32);
        }
    }
    if (lm == 0) {
        #pragma unroll
        for (int r = 0; r < 8; ++r) {
            int m = mrow + r;
            float inv = 1.0f / (aw[r] + 1e-8f);
            out[0 * MVOX + m] = ax[r] * inv;
            out[1 * MVOX + m] = ay[r] * inv;
            out[2 * MVOX + m] = az[r] * inv;
        }
    }
}

// ---------------------------------------------------------------------------
extern "C" void kernel_launch(void* const* d_in, const int* in_sizes, int n_in,
                              void* d_out, int out_size, void* d_ws, size_t ws_size,
                              hipStream_t stream)
{
    const float* kpts  = (const float*)d_in[0]; // (1,1024,3)
    const float* disp  = (const float*)d_in[1]; // (1,1024,3)
    const float* feats = (const float*)d_in[2]; // (1,16,48,48,48)
    const float* w0    = (const float*)d_in[3]; // (19,19,3,3,3)
    const float* b0    = (const float*)d_in[4]; // (19)
    const float* w1    = (const float*)d_in[5]; // (8,19,3,3,3)
    const float* b1    = (const float*)d_in[6]; // (8)
    float* out = (float*)d_out;                 // (1,3,48,48,48)

    float* ws  = (float*)d_ws;
    float* h0  = ws;                       // 19*MVOX
    float* h1  = h0 + (size_t)19 * MVOX;   // 8*MVOX
    float* st0 = h1 + (size_t)8 * MVOX;    // 64
    float* st1 = st0 + 64;                 // 64
    float* fgn = st1 + 64;                 // MVOX
    float* fk  = fgn + MVOX;               // 8*NKP
    float* fkn = fk + 8 * NKP;             // NKP

    const int VB = MVOX / 256;             // 432 blocks, exact

    conv0_relu_kernel<<<VB, 256, 0, stream>>>(feats, w0, b0, h0);
    in_stats_kernel<<<19, 256, 0, stream>>>(h0, st0);
    in_apply_kernel<<<(19 * MVOX) / 256, 256, 0, stream>>>(h0, st0);
    conv1_relu_kernel<<<VB, 256, 0, stream>>>(h0, w1, b1, h1);
    in_stats_kernel<<<8, 256, 0, stream>>>(h1, st1);
    in_apply_fgnorm_kernel<<<VB, 256, 0, stream>>>(h1, st1, fgn);
    sample_kernel<<<NKP / 256, 256, 0, stream>>>(h1, kpts, fk, fkn);
    // MVOX/16 waves = 6912; 8 waves/block -> 864 blocks, exact coverage
    rbf_wmma_kernel<<<(MVOX / 16) / 8, 256, 0, stream>>>(h1, fgn, fk, fkn, disp, out);
}